// Fusion_3968549782066
// MI455X (gfx1250) — compile-verified
//
#include <hip/hip_runtime.h>
#include <math.h>

typedef float v2f __attribute__((ext_vector_type(2)));
typedef float v8f __attribute__((ext_vector_type(8)));

#define EPSV 1e-5f

// ---------------------------------------------------------------------------
// 1x1 conv (pointwise GEMM): Y[b,co,p] = sum_ci W[co,ci] * X[b,ci,p] + bias[co]
// One wave computes a 16(co) x 16(pixel) tile, K-loop over Cin in steps of 4
// using V_WMMA_F32_16X16X4_F32.
// ---------------------------------------------------------------------------
__global__ void conv1x1_wmma(const float* __restrict__ X, const float* __restrict__ Wt,
                             const float* __restrict__ Bi, float* __restrict__ Y,
                             int Cin, int HW, int tilesTotal) {
    int wid = blockIdx.x * (blockDim.x >> 5) + (threadIdx.x >> 5);
    if (wid >= tilesTotal) return;            // wave-uniform: EXEC stays all-ones
    int lane = threadIdx.x & 31;
    int half = lane >> 4;                      // 0: lanes 0-15, 1: lanes 16-31
    int lm   = lane & 15;
    int nT   = HW >> 4;
    int perB = 4 * nT;
    int b    = wid / perB;
    int r    = wid % perB;
    int co0  = (r / nT) * 16;
    int n0   = (r % nT) * 16;

    const float* wp = Wt + (size_t)(co0 + lm) * Cin;          // A row (out channel)
    const float* xp = X + (size_t)b * Cin * HW + n0 + lm;     // B column (pixel)

    v8f acc = {};
    for (int k = 0; k < Cin; k += 4) {
        int c0 = k + 2 * half;
        v2f a, bv;
        a.x  = wp[c0];                     // A[M=lm, K=c0]
        a.y  = wp[c0 + 1];                 // A[M=lm, K=c0+1]
        bv.x = xp[(size_t)c0 * HW];        // B[K=c0,   N=lm]
        bv.y = xp[(size_t)(c0 + 1) * HW];  // B[K=c0+1, N=lm]
        acc = __builtin_amdgcn_wmma_f32_16x16x4_f32(false, a, false, bv,
                                                    (short)0, acc, false, false);
    }
    for (int rr = 0; rr < 8; ++rr) {
        int m = rr + 8 * half;             // C/D layout: VGPR rr -> rows rr / rr+8
        Y[((size_t)b * 64 + co0 + m) * HW + n0 + lm] = acc[rr] + Bi[co0 + m];
    }
}

// ---------------------------------------------------------------------------
// 3x3 VALID conv as tap-accumulated GEMM (M=64 out channels, K=64*9).
// ---------------------------------------------------------------------------
__global__ void conv3x3_wmma(const float* __restrict__ X, const float* __restrict__ W3,
                             const float* __restrict__ Bi, float* __restrict__ Y,
                             int Hin, int Win, int Hout, int Wout,
                             int nT, int tilesTotal) {
    int wid = blockIdx.x * (blockDim.x >> 5) + (threadIdx.x >> 5);
    if (wid >= tilesTotal) return;
    int lane = threadIdx.x & 31;
    int half = lane >> 4;
    int lm   = lane & 15;
    int outHW = Hout * Wout;
    int perB  = 4 * nT;
    int b   = wid / perB;
    int r   = wid % perB;
    int co0 = (r / nT) * 16;
    int n   = (r % nT) * 16 + lm;
    int nc  = (n < outHW) ? n : (outHW - 1);   // clamp so WMMA lanes read valid mem
    int oy  = nc / Wout, ox = nc % Wout;

    const float* Xb = X + (size_t)b * 64 * Hin * Win;
    const float* Wr = W3 + (size_t)(co0 + lm) * 576;   // per-out-channel 64*9 block

    v8f acc = {};
    for (int tap = 0; tap < 9; ++tap) {
        int dy = tap / 3, dx = tap % 3;
        const float* xp = Xb + (size_t)(oy + dy) * Win + (ox + dx);
        const float* wp = Wr + tap;                    // + ci*9 per K element
        for (int ci = 0; ci < 64; ci += 4) {
            int c0 = ci + 2 * half;
            v2f a, bv;
            a.x  = wp[(size_t)c0 * 9];
            a.y  = wp[(size_t)(c0 + 1) * 9];
            bv.x = xp[(size_t)c0 * Hin * Win];
            bv.y = xp[(size_t)(c0 + 1) * Hin * Win];
            acc = __builtin_amdgcn_wmma_f32_16x16x4_f32(false, a, false, bv,
                                                        (short)0, acc, false, false);
        }
    }
    if (n < outHW) {
        for (int rr = 0; rr < 8; ++rr) {
            int m = rr + 8 * half;
            Y[((size_t)b * 64 + co0 + m) * outHW + n] = acc[rr] + Bi[co0 + m];
        }
    }
}

// ---------------------------------------------------------------------------
// Per-channel BN stats: stats[2c] += sum, stats[2c+1] += sum of squares
// over (N,H,W) for channel c.
// ---------------------------------------------------------------------------
__global__ void bn_stats(const float* __restrict__ Y, float* __restrict__ stats,
                         int C, int HW, int N, int bpc) {
    int c   = blockIdx.x / bpc;
    int blk = blockIdx.x % bpc;
    int tid = threadIdx.x;
    size_t M = (size_t)N * HW;
    float s = 0.f, q = 0.f;
    for (size_t idx = (size_t)blk * 256 + tid; idx < M; idx += (size_t)bpc * 256) {
        int n = (int)(idx / HW);
        int i = (int)(idx % HW);
        float v = Y[((size_t)n * C + c) * HW + i];
        s += v; q += v * v;
    }
    __shared__ float ss[256], sq[256];
    ss[tid] = s; sq[tid] = q;
    __syncthreads();
    for (int o = 128; o > 0; o >>= 1) {
        if (tid < o) { ss[tid] += ss[tid + o]; sq[tid] += sq[tid + o]; }
        __syncthreads();
    }
    if (tid == 0) {
        atomicAdd(&stats[2 * c], ss[0]);
        atomicAdd(&stats[2 * c + 1], sq[0]);
    }
}

// ---------------------------------------------------------------------------
// In-place BN apply, optionally fused with bilinear (align_corners=False)
// 2x-upsample skip-add from `prev`.
// ---------------------------------------------------------------------------
__global__ void bn_apply(float* __restrict__ U, const float* __restrict__ stats,
                         const float* __restrict__ g, const float* __restrict__ be,
                         const float* __restrict__ prev,
                         int C, int H, int W, int Hp, int Wp, int N) {
    size_t total = (size_t)N * C * H * W;
    size_t idx = (size_t)blockIdx.x * blockDim.x + threadIdx.x;
    if (idx >= total) return;
    int w = (int)(idx % W);
    int h = (int)((idx / W) % H);
    int c = (int)((idx / ((size_t)W * H)) % C);
    int n = (int)(idx / ((size_t)W * H * C));

    float M = (float)N * H * W;
    float m = stats[2 * c] / M;
    float v = stats[2 * c + 1] / M - m * m;
    float val = g[c] * (U[idx] - m) * rsqrtf(v + EPSV) + be[c];

    if (prev) {
        float hy = fminf(fmaxf((h + 0.5f) * ((float)Hp / H) - 0.5f, 0.f), (float)(Hp - 1));
        float wx = fminf(fmaxf((w + 0.5f) * ((float)Wp / W) - 0.5f, 0.f), (float)(Wp - 1));
        int h0 = (int)floorf(hy), w0 = (int)floorf(wx);
        int h1 = min(h0 + 1, Hp - 1), w1 = min(w0 + 1, Wp - 1);
        float th = hy - h0, tw = wx - w0;
        const float* P = prev + ((size_t)n * C + c) * Hp * Wp;
        float top = P[(size_t)h0 * Wp + w0] * (1.f - tw) + P[(size_t)h0 * Wp + w1] * tw;
        float bot = P[(size_t)h1 * Wp + w0] * (1.f - tw) + P[(size_t)h1 * Wp + w1] * tw;
        val += top * (1.f - th) + bot * th;
    }
    U[idx] = val;
}

// ---------------------------------------------------------------------------
// Per-sample cross-correlation (batched GEMV; q staged channel-wise into LDS
// with GLOBAL_LOAD_ASYNC_TO_LDS_B32, tracked by ASYNCcnt):
// A[b,y,x] = sum_{c,i,j} Q[b,c,i,j] * S[b,c, y+i-ph, x+j-pw], zero-padded S.
// S is (16,64,126,126); output is (16,127,127).
// ---------------------------------------------------------------------------
__global__ void corr_kernel(const float* __restrict__ Q, const float* __restrict__ S,
                            float* __restrict__ A, int qh, int qw) {
    __shared__ float qs[960];                 // max 30*30 = 900 floats
    const int SH = 126, OW = 127, OHW = 127 * 127;
    int b   = blockIdx.x;
    int tid = threadIdx.x;
    int p   = blockIdx.y * 256 + tid;
    int pc  = (p < OHW) ? p : (OHW - 1);
    int oy  = pc / OW, ox = pc % OW;
    int ph = qh / 2, pw = qw / 2;
    int qsz = qh * qw;
    int ry0 = oy - ph; if (ry0 < 0) ry0 = 0; if (ry0 > SH - 1) ry0 = SH - 1;
    float acc = 0.f;
    for (int c = 0; c < 64; ++c) {
        __syncthreads();                      // all waves done reading previous qs
        // Async-stage this channel's query tile straight into LDS (no VGPR hop).
        const float* qc = Q + ((size_t)b * 64 + c) * qsz;
        for (int t = tid; t < qsz; t += 256) {
            unsigned lds_off = (unsigned)(size_t)&qs[t];   // low 32 bits = LDS byte addr
            const float* gp = qc + t;
            asm volatile("global_load_async_to_lds_b32 %0, %1, off"
                         :: "v"(lds_off), "v"(gp) : "memory");
        }
        asm volatile("s_wait_asynccnt 0x0" ::: "memory");
        __syncthreads();                      // everyone's async writes have landed

        const float* Sb = S + ((size_t)b * 64 + c) * SH * SH;
        // Pull the next channel's search-map rows toward the WGP caches.
        if (c < 63)
            __builtin_prefetch(Sb + (size_t)SH * SH + (size_t)ry0 * SH, 0, 1);

        for (int i = 0; i < qh; ++i) {
            int sy = oy + i - ph;
            if (sy < 0 || sy >= SH) continue;
            const float* row = Sb + (size_t)sy * SH;
            const float* qr  = qs + i * qw;
            for (int j = 0; j < qw; ++j) {
                int sx = ox + j - pw;
                if (sx < 0 || sx >= SH) continue;
                acc = fmaf(qr[j], row[sx], acc);
            }
        }
    }
    if (p < OHW) A[(size_t)b * OHW + p] = acc;
}

// Whole-tensor sum / sumsq for the correlation BN.
__global__ void corr_stats(const float* __restrict__ A, float* __restrict__ st,
                           int L, int nblk) {
    int tid = threadIdx.x;
    float s = 0.f, q = 0.f;
    for (int i = blockIdx.x * 256 + tid; i < L; i += nblk * 256) {
        float v = A[i]; s += v; q += v * v;
    }
    __shared__ float ss[256], sq[256];
    ss[tid] = s; sq[tid] = q;
    __syncthreads();
    for (int o = 128; o > 0; o >>= 1) {
        if (tid < o) { ss[tid] += ss[tid + o]; sq[tid] += sq[tid + o]; }
        __syncthreads();
    }
    if (tid == 0) { atomicAdd(&st[0], ss[0]); atomicAdd(&st[1], sq[0]); }
}

// ---------------------------------------------------------------------------
// Final: softmax(fusion_w), per-map BN, weighted sum, bilinear align_corners
// 127 -> 128 upsample. out: (16,1,128,128).
// ---------------------------------------------------------------------------
__global__ void final_kernel(const float* __restrict__ A1, const float* __restrict__ A2,
                             const float* __restrict__ A3, const float* __restrict__ st,
                             const float* __restrict__ cg, const float* __restrict__ cb,
                             const float* __restrict__ fw, float* __restrict__ out) {
    int idx = blockIdx.x * blockDim.x + threadIdx.x;
    if (idx >= 16 * 128 * 128) return;
    int ox = idx & 127, oy = (idx >> 7) & 127, n = idx >> 14;

    float f0 = fw[0], f1 = fw[1], f2 = fw[2];
    float mx = fmaxf(f0, fmaxf(f1, f2));
    float e0 = expf(f0 - mx), e1 = expf(f1 - mx), e2 = expf(f2 - mx);
    float inv = 1.f / (e0 + e1 + e2);
    float w0 = e0 * inv, w1 = e1 * inv, w2 = e2 * inv;

    const float M2 = 16.f * 127.f * 127.f;
    float m0 = st[0] / M2, v0 = st[1] / M2 - m0 * m0;
    float m1 = st[2] / M2, v1 = st[3] / M2 - m1 * m1;
    float m2 = st[4] / M2, v2 = st[5] / M2 - m2 * m2;
    float s0 = cg[0] * rsqrtf(v0 + EPSV);
    float s1 = cg[1] * rsqrtf(v1 + EPSV);
    float s2 = cg[2] * rsqrtf(v2 + EPSV);
    float b0 = cb[0], b1 = cb[1], b2 = cb[2];

    auto fuse = [&](int h, int w) -> float {
        size_t p = (size_t)n * 16129 + (size_t)h * 127 + w;
        return w0 * (s0 * (A1[p] - m0) + b0)
             + w1 * (s1 * (A2[p] - m1) + b1)
             + w2 * (s2 * (A3[p] - m2) + b2);
    };

    float sc = 126.f / 127.f;                 // (in-1)/(out-1), align_corners=True
    float hy = oy * sc, wx = ox * sc;
    int h0 = (int)floorf(hy), w0i = (int)floorf(wx);
    int h1 = min(h0 + 1, 126), w1i = min(w0i + 1, 126);
    float th = hy - h0, tw = wx - w0i;
    float top = fuse(h0, w0i) * (1.f - tw) + fuse(h0, w1i) * tw;
    float bot = fuse(h1, w0i) * (1.f - tw) + fuse(h1, w1i) * tw;
    out[idx] = top * (1.f - th) + bot * th;
}

// ---------------------------------------------------------------------------
extern "C" void kernel_launch(void* const* d_in, const int* in_sizes, int n_in,
                              void* d_out, int out_size, void* d_ws, size_t ws_size,
                              hipStream_t stream) {
    const float* sat1 = (const float*)d_in[0];
    const float* sat2 = (const float*)d_in[1];
    const float* sat3 = (const float*)d_in[2];
    const float* uav1 = (const float*)d_in[3];
    const float* uav2 = (const float*)d_in[4];
    const float* uav3 = (const float*)d_in[5];
    const float* c1w = (const float*)d_in[6],  *c1b = (const float*)d_in[7];
    const float* b1g = (const float*)d_in[8],  *b1b = (const float*)d_in[9];
    const float* c2w = (const float*)d_in[10], *c2b = (const float*)d_in[11];
    const float* b2g = (const float*)d_in[12], *b2b = (const float*)d_in[13];
    const float* c3w = (const float*)d_in[14], *c3b = (const float*)d_in[15];
    const float* b3g = (const float*)d_in[16], *b3b = (const float*)d_in[17];
    const float* cuw = (const float*)d_in[18], *cub = (const float*)d_in[19];
    const float* bug = (const float*)d_in[20], *bub = (const float*)d_in[21];
    const float* crg = (const float*)d_in[22], *crb = (const float*)d_in[23];
    const float* fw  = (const float*)d_in[24];

    float* p = (float*)d_ws;
    float* U1d = p; p += 16 * 64 * 8 * 8;
    float* U2d = p; p += 16 * 64 * 16 * 16;
    float* U3d = p; p += 16 * 64 * 32 * 32;
    float* U1s = p; p += 16 * 64 * 32 * 32;
    float* U2s = p; p += 16 * 64 * 64 * 64;
    float* U3s = p; p += 16 * 64 * 128 * 128;
    float* C1d = p; p += 16 * 64 * 6 * 6;
    float* C2d = p; p += 16 * 64 * 14 * 14;
    float* C3d = p; p += 16 * 64 * 30 * 30;
    float* C3s = p; p += 16 * 64 * 126 * 126;
    float* A1  = p; p += 16 * 127 * 127;
    float* A2  = p; p += 16 * 127 * 127;
    float* A3  = p; p += 16 * 127 * 127;
    float* bnst = p; p += 128;   // 64 channels x {sum, sumsq}
    float* cst  = p; p += 8;     // 3 maps x {sum, sumsq}

    // 1x1 conv + training-mode BN (+ bilinear x2 skip add), level by level
    auto run_level = [&](const float* X, int Cin, int H, int W,
                         const float* Wt, const float* Bi,
                         const float* G, const float* Be,
                         float* U, const float* prev, int Hp, int Wp) {
        int HW = H * W;
        int tiles = 16 * 4 * (HW / 16);
        conv1x1_wmma<<<dim3((tiles + 3) / 4), 128, 0, stream>>>(X, Wt, Bi, U, Cin, HW, tiles);
        hipMemsetAsync(bnst, 0, 128 * sizeof(float), stream);
        long long M = 16LL * HW;
        long long bl = M / 4096; if (bl < 1) bl = 1; if (bl > 64) bl = 64;
        int bpc = (int)bl;
        bn_stats<<<dim3(64 * bpc), 256, 0, stream>>>(U, bnst, 64, HW, 16, bpc);
        size_t tot = (size_t)16 * 64 * HW;
        bn_apply<<<dim3((unsigned)((tot + 255) / 256)), 256, 0, stream>>>(
            U, bnst, G, Be, prev, 64, H, W, Hp, Wp, 16);
    };

    // UAV top-down path (i=0 weights)
    run_level(uav3, 512, 8, 8,    c1w,           c1b,      b1g,      b1b,      U1d, nullptr, 8, 8);
    run_level(uav2, 256, 16, 16,  c2w,           c2b,      b2g,      b2b,      U2d, U1d, 8, 8);
    run_level(uav1, 128, 32, 32,  c3w,           c3b,      b3g,      b3b,      U3d, U2d, 16, 16);
    // SAT top-down path (i=1 weights)
    run_level(sat3, 512, 32, 32,  c1w + 64 * 512, c1b + 64, b1g + 64, b1b + 64, U1s, nullptr, 32, 32);
    run_level(sat2, 256, 64, 64,  c2w + 64 * 256, c2b + 64, b2g + 64, b2b + 64, U2s, U1s, 32, 32);
    run_level(sat1, 128, 128, 128, c3w + 64 * 128, c3b + 64, b3g + 64, b3b + 64, U3s, U2s, 64, 64);

    // 3x3 VALID conv + BN ("cu")
    auto run_cu = [&](const float* Xin, int Hin, int Win, int i, float* Cout) {
        int Hout = Hin - 2, Wout = Win - 2;
        int outHW = Hout * Wout;
        int nT = (outHW + 15) / 16;
        int tiles = 16 * 4 * nT;
        conv3x3_wmma<<<dim3((tiles + 3) / 4), 128, 0, stream>>>(
            Xin, cuw + (size_t)i * 64 * 64 * 9, cub + i * 64, Cout,
            Hin, Win, Hout, Wout, nT, tiles);
        hipMemsetAsync(bnst, 0, 128 * sizeof(float), stream);
        long long M = 16LL * outHW;
        long long bl = M / 4096; if (bl < 1) bl = 1; if (bl > 64) bl = 64;
        int bpc = (int)bl;
        bn_stats<<<dim3(64 * bpc), 256, 0, stream>>>(Cout, bnst, 64, outHW, 16, bpc);
        size_t tot = (size_t)16 * 64 * outHW;
        bn_apply<<<dim3((unsigned)((tot + 255) / 256)), 256, 0, stream>>>(
            Cout, bnst, bug + i * 64, bub + i * 64, nullptr, 64, Hout, Wout, 1, 1, 16);
    };

    run_cu(U1d, 8, 8, 0, C1d);
    run_cu(U2d, 16, 16, 1, C2d);
    run_cu(U3d, 32, 32, 2, C3d);
    run_cu(U3s, 128, 128, 3, C3s);

    // Correlations + global BN stats
    hipMemsetAsync(cst, 0, 8 * sizeof(float), stream);
    int OHW = 127 * 127;
    dim3 cgrid(16, (unsigned)((OHW + 255) / 256));
    corr_kernel<<<cgrid, 256, 0, stream>>>(C1d, C3s, A1, 6, 6);
    corr_stats<<<dim3(64), 256, 0, stream>>>(A1, cst + 0, 16 * OHW, 64);
    corr_kernel<<<cgrid, 256, 0, stream>>>(C2d, C3s, A2, 14, 14);
    corr_stats<<<dim3(64), 256, 0, stream>>>(A2, cst + 2, 16 * OHW, 64);
    corr_kernel<<<cgrid, 256, 0, stream>>>(C3d, C3s, A3, 30, 30);
    corr_stats<<<dim3(64), 256, 0, stream>>>(A3, cst + 4, 16 * OHW, 64);

    // Fusion + final align-corners bilinear to 128x128
    int total = 16 * 128 * 128;
    final_kernel<<<dim3((total + 255) / 256), 256, 0, stream>>>(
        A1, A2, A3, cst, crg, crb, fw, (float*)d_out);
}